// DeepCluster_72361609003069
// MI455X (gfx1250) — compile-verified
//
#include <hip/hip_runtime.h>

// ---------------------------------------------------------------------------
// DeepCluster for MI455X (gfx1250): 3 flat WMMA-f32 GEMMs + LN + cdist/medoid.
// Flat-GEMM view: M = 384*768 rows, shared weights; fp32 matrix path
// (V_WMMA_F32_16X16X4_F32) to keep the discrete medoid argmins faithful.
// ---------------------------------------------------------------------------

typedef float v2f __attribute__((ext_vector_type(2)));
typedef float v8f __attribute__((ext_vector_type(8)));

#define M_TOTAL   294912L      // 384 * 768 rows of the flat MLP
#define N_GROUPS  384          // B*AFTER_F = 64*6
#define LP        98           // FD*(L_TOK-1) tokens
#define KC        49           // clusters
#define KPAD      100          // K=98 padded to multiple of 4
#define OUT_ELEMS (50L * 384L * 768L)

// ---------------------------------------------------------------------------
// Kernel 1: build dataT[n*768+d][100] = data[n][t][d] transposed per n.
// data[n=a*64+b][t=f*49+l][d] = x[1+l][b*12+a*2+f][d]
// ---------------------------------------------------------------------------
__global__ __launch_bounds__(256)
void pack_dataT(const float* __restrict__ x, float* __restrict__ dataT) {
  __shared__ float T[LP][65];
  const int n  = blockIdx.x;              // 0..383   (a*64+b ordering)
  const int dt = blockIdx.y;              // 0..11
  const int a = n >> 6, b = n & 63;
  const int d0 = dt * 64;
  for (int idx = threadIdx.x; idx < LP * 64; idx += 256) {
    const int t = idx >> 6, dd = idx & 63;
    const int l = t % 49, f = t / 49;
    const int bt = b * 12 + a * 2 + f;
    T[t][dd] = x[((long)(1 + l) * 768 + bt) * 768 + d0 + dd];
  }
  __syncthreads();
  for (int idx = threadIdx.x; idx < 64 * KPAD; idx += 256) {
    const int dd = idx / KPAD, t = idx % KPAD;
    dataT[((long)(n * 768 + d0 + dd)) * KPAD + t] = (t < LP) ? T[t][dd] : 0.f;
  }
}

// ---------------------------------------------------------------------------
// Kernel 2: pack weights into fragment-ordered, zero-padded float2 buffer:
//   Bq[(kc*50 + 2*s + h)*Npad + col] = { B[kc*98+4s+2h][col], B[..+1][col] }
// so the GEMM B-fragment is a single unconditional global_load_b64.
// ---------------------------------------------------------------------------
__global__ __launch_bounds__(256)
void pack_Bq(const float* __restrict__ src, float* __restrict__ dst,
             int kchunks, int Nvalid, int Npad) {
  const int total = kchunks * 50 * Npad;
  for (int idx = blockIdx.x * 256 + threadIdx.x; idx < total;
       idx += gridDim.x * 256) {
    const int col = idx % Npad;
    const int row = idx / Npad;
    const int kc = row / 50, rr = row % 50;
    const int s = rr >> 1, h = rr & 1;
    const int c = 4 * s + 2 * h;
    float a0 = 0.f, a1 = 0.f;
    if (col < Nvalid) {
      if (c < LP)     a0 = src[(long)(kc * LP + c) * Nvalid + col];
      if (c + 1 < LP) a1 = src[(long)(kc * LP + c + 1) * Nvalid + col];
    }
    dst[2 * idx]     = a0;
    dst[2 * idx + 1] = a1;
  }
}

// ---------------------------------------------------------------------------
// Kernel 3: C[M][NVALID] = A[M][K] * B + bias  via V_WMMA_F32_16X16X4_F32.
// Block = 256 thr (8 waves); wave owns 16 rows.  A block staged in LDS
// As[128][100] (zero-padded K chunk of 98->100); B fragments come from the
// pre-packed Bq buffer.  Inner loop is branch-free: per K-step one
// ds_load_b64 (A frag, reused across the tile group) + GT x (global_load_b64
// + wmma).  GT divides NTILES so everything fully unrolls.
// Fragment layout (ISA 7.12.2, 32-bit), lane = 16*h + r:
//   A vgpr0/1 : A[row0+r][k+2h] / [k+2h+1]     B vgpr0/1 : B[k+2h][col]/[k+2h+1][col]
//   C vgpr g  : C[row0+g+8h][col]
// ---------------------------------------------------------------------------
template<int KCHUNKS, int NTILES, int GT, int NVALID>
__global__ __launch_bounds__(256)
void gemm_wmma(const float* __restrict__ A, int lda,
               const float* __restrict__ BqF,
               const float* __restrict__ bias,
               float* __restrict__ C, int ldc) {
  constexpr int NPAD = NTILES * 16;
  constexpr int NG   = NTILES / GT;
  __shared__ float As[128][KPAD];
  const v2f* __restrict__ Bq = (const v2f*)BqF;
  const int wave = threadIdx.x >> 5;
  const int lane = threadIdx.x & 31;
  const int h = lane >> 4;
  const int r = lane & 15;
  const long rowB = (long)blockIdx.x * 128;
  const long row0 = rowB + wave * 16;

  for (int tg = 0; tg < NG; ++tg) {
    const int t0 = tg * GT;
    v8f acc[GT];
#pragma unroll
    for (int g = 0; g < GT; ++g) acc[g] = (v8f){};

    for (int kc = 0; kc < KCHUNKS; ++kc) {
      const bool doStage = (KCHUNKS > 1) || (tg == 0);
      if (doStage) {
        if (!(tg == 0 && kc == 0)) __syncthreads();   // protect prior readers
        for (int i = threadIdx.x; i < 128 * KPAD; i += 256) {
          const int rr = i / KPAD, cc = i % KPAD;
          As[rr][cc] = (cc < LP) ? A[(rowB + rr) * (long)lda + kc * LP + cc]
                                 : 0.f;
        }
        __syncthreads();
      }
      const v2f* Brow = Bq + (long)kc * 50 * NPAD;
#pragma unroll
      for (int s = 0; s < 25; ++s) {
        const v2f a = *(const v2f*)&As[wave * 16 + r][4 * s + 2 * h];
        const v2f* bp = Brow + (2 * s + h) * NPAD + t0 * 16 + r;
#pragma unroll
        for (int g = 0; g < GT; ++g) {
          const v2f bb = bp[g * 16];
          acc[g] = __builtin_amdgcn_wmma_f32_16x16x4_f32(
              false, a, false, bb, (short)0, acc[g], false, false);
        }
      }
    }
    // epilogue: bias + masked store (pad columns computed but dropped)
#pragma unroll
    for (int g = 0; g < GT; ++g) {
      const int col = (t0 + g) * 16 + r;
      if (col < NVALID) {
        const float bi = bias[col];
        float* Cp = C + (row0 + 8 * h) * (long)ldc + col;
#pragma unroll
        for (int q = 0; q < 8; ++q)
          Cp[(long)q * ldc] = acc[g][q] + bi;
      }
    }
  }
}

// ---------------------------------------------------------------------------
// Kernel 4: row-wise LayerNorm (one wave per row); in==out allowed.
// ---------------------------------------------------------------------------
__global__ __launch_bounds__(256)
void layernorm_rows(const float* __restrict__ in, float* __restrict__ out,
                    const float* __restrict__ w, const float* __restrict__ b,
                    int Nvalid) {
  const long row = (long)blockIdx.x * 8 + (threadIdx.x >> 5);
  const int lane = threadIdx.x & 31;
  const float* x = in + row * (long)Nvalid;
  float* y = out + row * (long)Nvalid;
  float s = 0.f;
  for (int i = lane; i < Nvalid; i += 32) s += x[i];
#pragma unroll
  for (int off = 16; off > 0; off >>= 1) s += __shfl_xor(s, off, 32);
  const float mu = s / (float)Nvalid;
  float v = 0.f;
  for (int i = lane; i < Nvalid; i += 32) { float d = x[i] - mu; v += d * d; }
#pragma unroll
  for (int off = 16; off > 0; off >>= 1) v += __shfl_xor(v, off, 32);
  const float inv = 1.f / sqrtf(v / (float)Nvalid + 1e-5f);
  for (int i = lane; i < Nvalid; i += 32)
    y[i] = (x[i] - mu) * inv * w[i] + b[i];
}

// ---------------------------------------------------------------------------
// Kernel 5: per-n cdist + loss + medoid selection (one 128-thr block per n).
// Thread t<98 owns token row l=t with 49 register accumulators; centroid
// slab reads broadcast across lanes.  Medoid rule is the shift-free
// equivalent of the reference (unassigned clusters -> index 0).
// ---------------------------------------------------------------------------
__global__ __launch_bounds__(128)
void cdist_medoid(const float* __restrict__ dataT, const float* __restrict__ cent,
                  float* __restrict__ lossBuf, int* __restrict__ medoids) {
  __shared__ float Asl[16][KPAD];
  __shared__ float Csl[16][KC];
  __shared__ float sqCs[KC];
  __shared__ float cdS[LP];
  __shared__ int   asS[LP];
  __shared__ int   medS[KC];
  const int n = blockIdx.x;
  const int t = threadIdx.x;
  const float* Ab = dataT + (long)n * 768 * KPAD;
  const float* Cb = cent  + (long)n * 768 * KC;

  float acc[KC];
#pragma unroll
  for (int k = 0; k < KC; ++k) acc[k] = 0.f;
  float sqa = 0.f, sqc = 0.f;

  for (int c = 0; c < 48; ++c) {                 // 48 * 16 = 768 d-slabs
    const int d0 = c * 16;
    for (int i = t; i < 16 * KPAD; i += 128)
      Asl[i / KPAD][i % KPAD] = Ab[(long)(d0 + i / KPAD) * KPAD + i % KPAD];
    for (int i = t; i < 16 * KC; i += 128)
      Csl[i / KC][i % KC] = Cb[(long)(d0 + i / KC) * KC + i % KC];
    __syncthreads();
    if (t < LP) {
      for (int dd = 0; dd < 16; ++dd) {
        const float a = Asl[dd][t];
        sqa += a * a;
#pragma unroll
        for (int k = 0; k < KC; ++k) acc[k] += a * Csl[dd][k];
      }
    }
    if (t < KC) {
      for (int dd = 0; dd < 16; ++dd) { const float cc = Csl[dd][t]; sqc += cc * cc; }
    }
    __syncthreads();
  }

  if (t < KC) sqCs[t] = sqc;
  __syncthreads();

  if (t < LP) {
    float best = 3.4e38f; int bi = 0;
#pragma unroll
    for (int k = 0; k < KC; ++k) {
      const float d2 = sqa + sqCs[k] - 2.f * acc[k];
      const float dv = sqrtf(fmaxf(d2, 0.f));
      if (dv < best) { best = dv; bi = k; }      // strict < keeps first index
    }
    cdS[t] = best; asS[t] = bi;
  }
  __syncthreads();

  if (t < KC) {
    float best = 3.4e38f; int bi = -1;
    for (int l = 0; l < LP; ++l)
      if (asS[l] == t && cdS[l] < best) { best = cdS[l]; bi = l; }
    medS[t] = (bi < 0) ? 0 : bi;
  }
  __syncthreads();

  if (t == 0) {
    float s = 0.f;
    for (int l = 0; l < LP; ++l) s += cdS[l];    // fixed order -> deterministic
    lossBuf[n] = s;
    for (int i = 1; i < KC; ++i) {               // insertion sort (jnp.sort)
      const int key = medS[i]; int j = i - 1;
      while (j >= 0 && medS[j] > key) { medS[j + 1] = medS[j]; --j; }
      medS[j + 1] = key;
    }
    for (int k = 0; k < KC; ++k) medoids[n * KC + k] = medS[k];
  }
}

// ---------------------------------------------------------------------------
// Kernel 6: assemble out[50][384][768].  nb = b*6+a ordering; medoids in
// n' = a*64+b ordering.  Medoid rows are contiguous 768-float rows of x.
// ---------------------------------------------------------------------------
__global__ __launch_bounds__(256)
void write_out(const float* __restrict__ x, const int* __restrict__ medoids,
               float* __restrict__ out) {
  const int nb  = blockIdx.x;                    // 0..383 (b*6+a)
  const int tkn = blockIdx.y;                    // 0..49
  const int b = nb / 6, a = nb % 6;
  float* dst = out + ((long)tkn * 384 + nb) * 768;
  if (tkn == 0) {
    const int bt0 = b * 12 + a * 2;
    const float* s0 = x + (long)bt0 * 768;       // x[0][bt0][*]
    const float* s1 = s0 + 768;                  // x[0][bt0+1][*]
    for (int i = threadIdx.x; i < 768; i += 256)
      dst[i] = 0.5f * (s0[i] + s1[i]);
  } else {
    const int k = tkn - 1;
    const int np = a * 64 + b;
    const int m = medoids[np * KC + k];
    const int l = m % 49, f = m / 49;
    const int bt = b * 12 + a * 2 + f;
    const float* s = x + ((long)(1 + l) * 768 + bt) * 768;
    for (int i = threadIdx.x; i < 768; i += 256)
      dst[i] = s[i];
  }
}

__global__ void loss_final(const float* __restrict__ lossBuf,
                           float* __restrict__ dst) {
  float s = 0.f;
  for (int n = 0; n < N_GROUPS; ++n) s += lossBuf[n];
  dst[0] = s / (float)N_GROUPS;
}

// ---------------------------------------------------------------------------
extern "C" void kernel_launch(void* const* d_in, const int* in_sizes, int n_in,
                              void* d_out, int out_size, void* d_ws, size_t ws_size,
                              hipStream_t stream) {
  (void)in_sizes; (void)n_in; (void)out_size; (void)ws_size;
  const float* x     = (const float*)d_in[0];
  const float* fc1_w = (const float*)d_in[1];
  const float* fc1_b = (const float*)d_in[2];
  const float* ln1_w = (const float*)d_in[3];
  const float* ln1_b = (const float*)d_in[4];
  const float* fc2_w = (const float*)d_in[5];
  const float* fc2_b = (const float*)d_in[6];
  const float* ln2_w = (const float*)d_in[7];
  const float* ln2_b = (const float*)d_in[8];
  const float* fc3_w = (const float*)d_in[9];
  const float* fc3_b = (const float*)d_in[10];
  const float* ln3_w = (const float*)d_in[11];
  const float* ln3_b = (const float*)d_in[12];
  float* out = (float*)d_out;

  const long M  = M_TOTAL;          // 294912
  const long CH = 8;
  const long Mc = M / CH;           // 36864 rows per chunk

  // workspace layout (floats), ~248.5 MB total; all 8B-aligned
  float* dataT = (float*)d_ws;                 // M * 100
  float* z1buf = dataT + M * KPAD;             // Mc * 392
  float* z2buf = z1buf + Mc * 392;             // Mc * 98
  float* z3    = z2buf + Mc * LP;              // M * 49
  float* lossB = z3    + M * KC;               // 384
  int*   med   = (int*)(lossB + N_GROUPS);     // 384 * 49
  float* Bq1   = (float*)(med + N_GROUPS * KC);// 1*50*400*2 = 40000
  float* Bq2   = Bq1 + 40000;                  // 4*50*112*2 = 44800
  float* Bq3   = Bq2 + 44800;                  // 1*50*64*2  = 6400

  // 0) pack weights into fragment-ordered padded buffers
  pack_Bq<<<79, 256, 0, stream>>>(fc1_w, Bq1, 1, 392, 400);
  pack_Bq<<<88, 256, 0, stream>>>(fc2_w, Bq2, 4, LP, 112);
  pack_Bq<<<13, 256, 0, stream>>>(fc3_w, Bq3, 1, KC, 64);

  // 1) pack x -> dataT
  pack_dataT<<<dim3(N_GROUPS, 12), 256, 0, stream>>>(x, dataT);

  // 2) MLP: three WMMA GEMMs + layernorms, chunked over M to bound workspace
  for (long c = 0; c < CH; ++c) {
    const long r0 = c * Mc;
    // z1 = LN1(dataT @ fc1_w + b1)   K=98(pad100), N=392 (25 tiles, groups of 5)
    gemm_wmma<1, 25, 5, 392><<<Mc / 128, 256, 0, stream>>>(
        dataT + r0 * KPAD, KPAD, Bq1, fc1_b, z1buf, 392);
    layernorm_rows<<<Mc / 8, 256, 0, stream>>>(z1buf, z1buf, ln1_w, ln1_b, 392);
    // z2 = LN2(z1 @ fc2_w + b2)      K=392 (4 chunks of 98->100), N=98 (7 tiles)
    gemm_wmma<4, 7, 7, LP><<<Mc / 128, 256, 0, stream>>>(
        z1buf, 392, Bq2, fc2_b, z2buf, LP);
    layernorm_rows<<<Mc / 8, 256, 0, stream>>>(z2buf, z2buf, ln2_w, ln2_b, LP);
    // z3 = z2 @ fc3_w + b3           K=98(pad100), N=49 (4 tiles)
    gemm_wmma<1, 4, 4, KC><<<Mc / 128, 256, 0, stream>>>(
        z2buf, LP, Bq3, fc3_b, z3 + r0 * KC, KC);
  }
  // 3) LN3 -> centroids (stored [row=(n,d)][k])
  layernorm_rows<<<M / 8, 256, 0, stream>>>(z3, z3, ln3_w, ln3_b, KC);

  // 4) per-n distances, loss, medoids
  cdist_medoid<<<N_GROUPS, 128, 0, stream>>>(dataT, z3, lossB, med);

  // 5) output assembly + deterministic loss reduction
  write_out<<<dim3(N_GROUPS, 50), 256, 0, stream>>>(x, med, out);
  loss_final<<<1, 1, 0, stream>>>(lossB, out + OUT_ELEMS);
}